// TCFPEmbedding_47132971107083
// MI455X (gfx1250) — compile-verified
//
#include <hip/hip_runtime.h>

// TCFP8 fake-quantized embedding gather for MI455X (gfx1250, wave32).
//
// Reference grid == OCP FP8 E4M3 exactly (e in [-6,8], step 2^(e-3), RNE,
// clamp +-448, subnormal steps 2^-9), so the quantizer is a native CDNA5
// FP8 round-trip: v_cvt_pk_fp8_f32 -> v_cvt_pk_f32_fp8.
//
// Layout: 1 row (1024 fp32) per workgroup of 64 threads; 16 floats/thread
// (4x b128 in one clause). A 32-float quant block spans 2 lanes -> absmax =
// in-lane max tree (v_max3-foldable) + 1 wave32 xor-shuffle. One division
// (448/amax) amortized over 16 elements.
//
// Bandwidth-bound: ~268 MB traffic => ~11.5 us floor at 23.3 TB/s; issue
// traffic ~8.5M wave-instructions -> well under the memory floor.

typedef float v4f __attribute__((ext_vector_type(4)));
typedef float v2f __attribute__((ext_vector_type(2)));

#define DIM      1024
#define EPT      16            // elements per thread
#define THREADS  (DIM / EPT)   // 64 = 2 wave32 waves per workgroup
#define MAXV     448.0f        // E4M3 max
#define INV_MAXV (1.0f / 448.0f)

#if __has_builtin(__builtin_amdgcn_cvt_pk_fp8_f32) && \
    __has_builtin(__builtin_amdgcn_cvt_pk_f32_fp8)
#define USE_HW_FP8 1
#else
#define USE_HW_FP8 0
#endif

#if !USE_HW_FP8
// Fallback: manual e4m3 grid rounding (exact via exponent-field extract).
__device__ __forceinline__ float fq_e4m3_sw(float v) {
    float av = __builtin_fabsf(v);
    int e = (int)(__float_as_uint(av) >> 23) - 127;
    e = e < -6 ? -6 : (e > 8 ? 8 : e);
    float step     = __int_as_float((e + 124) << 23); // 2^(e-3)
    float inv_step = __int_as_float((130 - e) << 23); // 2^(3-e)
    float q = __builtin_rintf(v * inv_step) * step;
    q = fminf(fmaxf(q, -MAXV), MAXV);
    return (av > 0.0f) ? q : 0.0f;
}
#endif

__device__ __forceinline__ float amax4(v4f w) {
    return fmaxf(fmaxf(__builtin_fabsf(w.x), __builtin_fabsf(w.y)),
                 fmaxf(__builtin_fabsf(w.z), __builtin_fabsf(w.w)));
}

// fp8 round-trip of one float2 pair is folded into 4-wide packs below.

__global__ __launch_bounds__(THREADS)
void tcfp8_embed_gather(const float* __restrict__ weight,
                        const int*   __restrict__ indices,
                        float*       __restrict__ out,
                        int n_rows) {
    const int row = blockIdx.x;
    const int tid = threadIdx.x;          // 0..63, 16 consecutive floats each

    const int idx = indices[row];         // uniform -> s_load
    const v4f* __restrict__ wrow = (const v4f*)(weight + (size_t)idx * DIM);
    v4f*       __restrict__ orow = (v4f*)(out + (size_t)row * DIM);

    // CDNA5 prefetch: warm L2 with the next row's 4 KB (one lane / 128 B line).
    if ((tid & 1) == 0 && row + 1 < n_rows) {
        const int nidx = indices[row + 1];
        const char* np = (const char*)(weight + (size_t)nidx * DIM);
        __builtin_prefetch(np + (size_t)tid * 64, /*rw=*/0, /*locality=*/1);
    }

    // Four b128 loads in one clause: this thread's 16 floats (one half-block).
    v4f w0 = wrow[4 * tid + 0];
    v4f w1 = wrow[4 * tid + 1];
    v4f w2 = wrow[4 * tid + 2];
    v4f w3 = wrow[4 * tid + 3];

    // Blockwise absmax over 32 floats = 2 lanes x 16 elems.
    float ax = fmaxf(fmaxf(amax4(w0), amax4(w1)),
                     fmaxf(amax4(w2), amax4(w3)));
    ax = fmaxf(ax, __shfl_xor(ax, 1, 32));

    const bool  nz        = ax > 0.0f;
    const float scale     = nz ? ax * INV_MAXV : 1.0f;
    const float inv_scale = nz ? MAXV / ax     : 1.0f;

    // Normalize into [-448,448]; pairs -> v_pk_mul_f32.
    v2f a0 = {w0.x, w0.y}; v2f a1 = {w0.z, w0.w};
    v2f a2 = {w1.x, w1.y}; v2f a3 = {w1.z, w1.w};
    v2f a4 = {w2.x, w2.y}; v2f a5 = {w2.z, w2.w};
    v2f a6 = {w3.x, w3.y}; v2f a7 = {w3.z, w3.w};
    a0 *= inv_scale; a1 *= inv_scale; a2 *= inv_scale; a3 *= inv_scale;
    a4 *= inv_scale; a5 *= inv_scale; a6 *= inv_scale; a7 *= inv_scale;

#if USE_HW_FP8
    // Native E4M3 round-trip: RNE onto the fp8 grid, back to f32.
    int p0 = __builtin_amdgcn_cvt_pk_fp8_f32(a0.x, a0.y, 0,  false);
    p0     = __builtin_amdgcn_cvt_pk_fp8_f32(a1.x, a1.y, p0, true);
    int p1 = __builtin_amdgcn_cvt_pk_fp8_f32(a2.x, a2.y, 0,  false);
    p1     = __builtin_amdgcn_cvt_pk_fp8_f32(a3.x, a3.y, p1, true);
    int p2 = __builtin_amdgcn_cvt_pk_fp8_f32(a4.x, a4.y, 0,  false);
    p2     = __builtin_amdgcn_cvt_pk_fp8_f32(a5.x, a5.y, p2, true);
    int p3 = __builtin_amdgcn_cvt_pk_fp8_f32(a6.x, a6.y, 0,  false);
    p3     = __builtin_amdgcn_cvt_pk_fp8_f32(a7.x, a7.y, p3, true);
    v2f q0 = __builtin_amdgcn_cvt_pk_f32_fp8(p0, false);
    v2f q1 = __builtin_amdgcn_cvt_pk_f32_fp8(p0, true);
    v2f q2 = __builtin_amdgcn_cvt_pk_f32_fp8(p1, false);
    v2f q3 = __builtin_amdgcn_cvt_pk_f32_fp8(p1, true);
    v2f q4 = __builtin_amdgcn_cvt_pk_f32_fp8(p2, false);
    v2f q5 = __builtin_amdgcn_cvt_pk_f32_fp8(p2, true);
    v2f q6 = __builtin_amdgcn_cvt_pk_f32_fp8(p3, false);
    v2f q7 = __builtin_amdgcn_cvt_pk_f32_fp8(p3, true);
#else
    v2f q0 = {fq_e4m3_sw(a0.x), fq_e4m3_sw(a0.y)};
    v2f q1 = {fq_e4m3_sw(a1.x), fq_e4m3_sw(a1.y)};
    v2f q2 = {fq_e4m3_sw(a2.x), fq_e4m3_sw(a2.y)};
    v2f q3 = {fq_e4m3_sw(a3.x), fq_e4m3_sw(a3.y)};
    v2f q4 = {fq_e4m3_sw(a4.x), fq_e4m3_sw(a4.y)};
    v2f q5 = {fq_e4m3_sw(a5.x), fq_e4m3_sw(a5.y)};
    v2f q6 = {fq_e4m3_sw(a6.x), fq_e4m3_sw(a6.y)};
    v2f q7 = {fq_e4m3_sw(a7.x), fq_e4m3_sw(a7.y)};
#endif

    // Rescale; pairs -> v_pk_mul_f32.
    q0 *= scale; q1 *= scale; q2 *= scale; q3 *= scale;
    q4 *= scale; q5 *= scale; q6 *= scale; q7 *= scale;

    v4f r0 = {q0.x, q0.y, q1.x, q1.y};
    v4f r1 = {q2.x, q2.y, q3.x, q3.y};
    v4f r2 = {q4.x, q4.y, q5.x, q5.y};
    v4f r3 = {q6.x, q6.y, q7.x, q7.y};

    // Write-only 134 MB stream: non-temporal b128 stores, keep L2 for weights.
    __builtin_nontemporal_store(r0, &orow[4 * tid + 0]);
    __builtin_nontemporal_store(r1, &orow[4 * tid + 1]);
    __builtin_nontemporal_store(r2, &orow[4 * tid + 2]);
    __builtin_nontemporal_store(r3, &orow[4 * tid + 3]);
}

extern "C" void kernel_launch(void* const* d_in, const int* in_sizes, int n_in,
                              void* d_out, int out_size, void* d_ws, size_t ws_size,
                              hipStream_t stream) {
    const float* weight  = (const float*)d_in[0];
    const int*   indices = (const int*)d_in[1];   // harness passes integer inputs as int32
    float*       out     = (float*)d_out;

    const int n_rows = in_sizes[1];               // 8 * 4096 = 32768 gathered rows

    dim3 grid((unsigned)n_rows), block(THREADS);
    tcfp8_embed_gather<<<grid, block, 0, stream>>>(weight, indices, out, n_rows);
}